// TemporalAggregator_32547262169236
// MI455X (gfx1250) — compile-verified
//
#include <hip/hip_runtime.h>

#define N_IN_DIM 256
#define N_HID    128
#define N_HEADS  4
#define HEAD_DIM 32
#define EDGE_CAP 1024

typedef float v2f __attribute__((ext_vector_type(2)));
typedef float v8f __attribute__((ext_vector_type(8)));
typedef unsigned int u32x4 __attribute__((ext_vector_type(4)));
typedef int i32x4 __attribute__((ext_vector_type(4)));
typedef int i32x8 __attribute__((ext_vector_type(8)));

#if defined(__has_builtin)
#if __has_builtin(__builtin_amdgcn_tensor_load_to_lds)
#define TGA_HAVE_TDM 1
#endif
#endif

// ---------------------------------------------------------------------------
// Kernel 1: tf = node_features @ Wf^T + bf   (M x 256) * (256 x 128)
// WMMA f32 16x16x4, one wave -> 16 rows x 128 cols, 8 f32 accumulators.
// Wf staged into LDS in K-chunks of 64. Row stride 68 floats -> the 16-lane
// B-fragment reads land on distinct banks ((n*68+k) % 64 distinct).
// Staging uses the Tensor Data Mover: one D#-descriptor DMA per chunk with
// hardware LDS padding (interval 64 DWORDs, amount 4 DWORDs), issued by wave 0
// and tracked with TENSORcnt. Fallback: cooperative float4 loads.
// ---------------------------------------------------------------------------
__global__ __launch_bounds__(256) void tga_tf_gemm(
    const float* __restrict__ A,    // [M, 256]
    const float* __restrict__ Wf,   // [128, 256] row-major
    const float* __restrict__ bf,   // [128]
    float* __restrict__ tf,         // [M, 128]
    int M)
{
    __shared__ __align__(16) float sWf[N_HID][68];   // 64-wide K chunk + 4 pad

    const int wave = threadIdx.x >> 5;
    const int lane = threadIdx.x & 31;
    const bool hi  = lane >= 16;
    const int l16  = lane & 15;

    const int row0  = (blockIdx.x * 8 + wave) * 16;        // wave's 16-row stripe
    const int m     = row0 + l16;
    const int mload = m < M ? m : (M - 1);                 // clamp, keep EXEC full
    const float* Arow = A + (size_t)mload * N_IN_DIM;

    v8f acc[8] = {};

#if TGA_HAVE_TDM
    const unsigned ldsOff = (unsigned)(unsigned long long)(const void*)&sWf[0][0];
#endif

    for (int kc = 0; kc < N_IN_DIM; kc += 64) {
#if TGA_HAVE_TDM
        if (wave == 0) {
            const unsigned long long ga =
                (unsigned long long)(const void*)Wf + (unsigned long long)kc * 4ull;
            // D# group 0: count=1 | lds_addr | global_addr | type=2
            u32x4 g0 = { 1u,
                         ldsOff,
                         (unsigned)(ga & 0xffffffffull),
                         ((unsigned)((ga >> 32) & 0x01ffffffull)) | 0x80000000u };
            // D# group 1:
            //  dw0: data_size=4B(code2), pad_enable=1, pad_interval=64DW(code5),
            //       pad_amount=4DW(code3), workgroup_mask=0
            //  tensor_dim0=256, tensor_dim1=128, tile_dim0=64, tile_dim1=128,
            //  tensor_dim0_stride=256
            i32x8 g1 = { 0x07520000,
                         (256 << 16),
                         (128 << 16),
                         (64  << 16),
                         128,
                         256,
                         0,
                         0 };
            i32x4 gz = { 0, 0, 0, 0 };
#if defined(__clang_major__) && (__clang_major__ >= 23)
            i32x8 gz8 = { 0, 0, 0, 0, 0, 0, 0, 0 };
            __builtin_amdgcn_tensor_load_to_lds(g0, g1, gz, gz, gz8, 0);
#else
            __builtin_amdgcn_tensor_load_to_lds(g0, g1, gz, gz, 0);
#endif
            __builtin_amdgcn_s_wait_tensorcnt(0);
        }
        __syncthreads();
#else
        // cooperative stage: 128 rows x 64 floats = 2048 float4
        for (int i = threadIdx.x; i < N_HID * 16; i += 256) {
            const int n  = i >> 4;
            const int k4 = i & 15;
            const float4 v = *(const float4*)&Wf[n * N_IN_DIM + kc + k4 * 4];
            *(float4*)&sWf[n][k4 * 4] = v;
        }
        __syncthreads();
#endif
        if (kc + 64 < N_IN_DIM) __builtin_prefetch(&Arow[kc + 64], 0, 0);

        for (int k = 0; k < 64; k += 4) {
            const int kk = k + (hi ? 2 : 0);
            v2f a;
            a.x = Arow[kc + kk];
            a.y = Arow[kc + kk + 1];
#pragma unroll
            for (int t = 0; t < 8; ++t) {
                const int n = t * 16 + l16;
                v2f bfrag;
                bfrag.x = sWf[n][kk];        // B[k][n]   = Wf[n][k]
                bfrag.y = sWf[n][kk + 1];    // B[k+1][n]
                acc[t] = __builtin_amdgcn_wmma_f32_16x16x4_f32(
                    false, a, false, bfrag, (short)0, acc[t], false, false);
            }
        }
        __syncthreads();
    }

    // D layout: lane (0-15|16-31) -> col n = t*16 + l16 ; VGPR r -> row r (+8 hi)
#pragma unroll
    for (int t = 0; t < 8; ++t) {
        const int n   = t * 16 + l16;
        const float bias = bf[n];
#pragma unroll
        for (int r = 0; r < 8; ++r) {
            const int mm = row0 + r + (hi ? 8 : 0);
            if (mm < M) tf[(size_t)mm * N_HID + n] = acc[t][r] + bias;
        }
    }
}

// ---------------------------------------------------------------------------
// Kernel 2: deterministic incident-edge scan. One block per target, 8 waves,
// each wave owns a contiguous edge chunk. Phase 1 counts, LDS prefix over
// waves, phase 2 ballot-compacts in order. No atomics -> replay-stable order.
// bit31 of stored edge id: 0 = src==tgt (use tf[dst]), 1 = dst-only (tf[src]).
// ---------------------------------------------------------------------------
__global__ __launch_bounds__(256) void tga_scan(
    const int* __restrict__ src, const int* __restrict__ dst,
    const int* __restrict__ tgt, int E,
    int* __restrict__ counts, int* __restrict__ elist)
{
    const int b = blockIdx.x;
    const int t = tgt[b];
    const int wave = threadIdx.x >> 5;
    const int lane = threadIdx.x & 31;

    const int per = (E + 7) / 8;
    const int beg = wave * per;
    const int end = min(beg + per, E);

    __shared__ int wcnt[8];
    __shared__ int woff[9];

    int cnt = 0;
    for (int base = beg; base < end; base += 32) {
        const int e = base + lane;
        if (e < end && (src[e] == t || dst[e] == t)) cnt++;
    }
    for (int o = 16; o; o >>= 1) cnt += __shfl_xor(cnt, o);
    if (lane == 0) wcnt[wave] = cnt;
    __syncthreads();
    if (threadIdx.x == 0) {
        int s = 0;
        for (int w = 0; w < 8; ++w) { woff[w] = s; s += wcnt[w]; }
        woff[8] = s;
        counts[b] = s < EDGE_CAP ? s : EDGE_CAP;
    }
    __syncthreads();

    int pos = woff[wave];
    for (int base = beg; base < end; base += 32) {
        const int e = base + lane;
        const bool isS = (e < end) && (src[e] == t);
        const bool hit = (e < end) && (isS || dst[e] == t);
        const unsigned mask = (unsigned)__ballot(hit);          // wave32
        if (hit) {
            const int p = pos + __popc(mask & ((1u << lane) - 1u));
            if (p < EDGE_CAP)
                elist[b * EDGE_CAP + p] = isS ? e : (e | 0x80000000);
        }
        pos += __popc(mask);
    }
}

// ---------------------------------------------------------------------------
// Kernel 3: per-target attention. 128 threads = 4 waves, wave w == head w.
// Online (flash) softmax over the incident edge list; exact match to the
// reference since masked lanes underflow to p==0 there.
// ---------------------------------------------------------------------------
__global__ __launch_bounds__(128) void tga_attn(
    const float* __restrict__ tf,
    const int* __restrict__ tgt,
    const int* __restrict__ src, const int* __restrict__ dst,
    const float* __restrict__ ts,
    const float* __restrict__ Wt1, const float* __restrict__ bt1,
    const float* __restrict__ Wt2, const float* __restrict__ bt2,
    const float* __restrict__ Wq,  const float* __restrict__ bq,
    const float* __restrict__ Wk,  const float* __restrict__ bk,
    const float* __restrict__ Wv,  const float* __restrict__ bv,
    const float* __restrict__ W_out, const float* __restrict__ b_out,
    const float* __restrict__ Wo,  const float* __restrict__ bo,
    const int* __restrict__ counts, const int* __restrict__ elist,
    float* __restrict__ out)
{
    const int b    = blockIdx.x;
    const int tid  = threadIdx.x;          // 0..127, owns output dim `tid`
    const float scale = 0.1767766952966369f;   // 1/sqrt(32)

    __shared__ float s_tf[N_HID];
    __shared__ float s_q[N_HID];
    __shared__ float s_h1[N_HID];
    __shared__ float s_key[N_HID];
    __shared__ float s_ctx[N_HID];
    __shared__ float s_agg[N_HID];

    const int node = tgt[b];
    s_tf[tid] = tf[(size_t)node * N_HID + tid];
    __syncthreads();

    float acc = bq[tid];
    for (int j = 0; j < N_HID; ++j) acc += s_tf[j] * Wq[tid * N_HID + j];
    s_q[tid] = acc * scale;
    __syncthreads();

    const float q = s_q[tid];
    float m = -INFINITY, l = 0.f, ctx = 0.f;
    const int cnt = counts[b];

    for (int i = 0; i < cnt; ++i) {
        const int ee   = elist[b * EDGE_CAP + i];
        const int e    = ee & 0x7fffffff;
        const bool isS = ee >= 0;
        const int nb   = isS ? dst[e] : src[e];
        const float te = ts[e];

        s_h1[tid] = fmaxf(te * Wt1[tid] + bt1[tid], 0.f);
        __syncthreads();

        float tfeat = bt2[tid];
        for (int j = 0; j < N_HID; ++j) tfeat += s_h1[j] * Wt2[tid * N_HID + j];
        s_key[tid] = tf[(size_t)nb * N_HID + tid] + tfeat;
        __syncthreads();

        float k = bk[tid], v = bv[tid];
        for (int j = 0; j < N_HID; ++j) {
            const float kj = s_key[j];
            k += kj * Wk[tid * N_HID + j];
            v += kj * Wv[tid * N_HID + j];
        }

        float s = q * k;                               // per-head dot over wave
        for (int o = 16; o; o >>= 1) s += __shfl_xor(s, o);

        const float newm  = fmaxf(m, s);
        const float alpha = __expf(m - newm);
        const float w     = __expf(s - newm);
        l   = l * alpha + w;
        ctx = ctx * alpha + w * v;
        m   = newm;
        __syncthreads();                               // before s_h1/s_key reuse
    }

    s_ctx[tid] = (l > 0.f) ? ctx / l : 0.f;
    __syncthreads();

    float attn = b_out[tid];
    for (int j = 0; j < N_HID; ++j) attn += s_ctx[j] * W_out[tid * N_HID + j];
    s_agg[tid] = (cnt > 0) ? attn : s_tf[tid];
    __syncthreads();

    float o = bo[tid];
    for (int j = 0; j < N_HID; ++j) o += s_agg[j] * Wo[tid * N_HID + j];
    out[b * N_HID + tid] = o;
}

// ---------------------------------------------------------------------------
extern "C" void kernel_launch(void* const* d_in, const int* in_sizes, int n_in,
                              void* d_out, int out_size, void* d_ws, size_t ws_size,
                              hipStream_t stream) {
    const float* node_features = (const float*)d_in[0];
    const int*   edge_index    = (const int*)  d_in[1];
    const float* edge_ts       = (const float*)d_in[2];
    const int*   target_nodes  = (const int*)  d_in[3];
    const float* Wf   = (const float*)d_in[4];
    const float* bf   = (const float*)d_in[5];
    const float* Wt1  = (const float*)d_in[6];
    const float* bt1  = (const float*)d_in[7];
    const float* Wt2  = (const float*)d_in[8];
    const float* bt2  = (const float*)d_in[9];
    const float* W_in = (const float*)d_in[10];
    const float* b_in = (const float*)d_in[11];
    const float* W_out= (const float*)d_in[12];
    const float* b_out= (const float*)d_in[13];
    const float* Wo   = (const float*)d_in[14];
    const float* bo   = (const float*)d_in[15];

    const int E = in_sizes[2];
    const int B = in_sizes[3];
    const int N = in_sizes[0] / N_IN_DIM;

    // workspace layout
    float* tf = (float*)d_ws;
    size_t off = ((size_t)N * N_HID * sizeof(float) + 255) & ~(size_t)255;
    int* counts = (int*)((char*)d_ws + off);
    off += ((size_t)B * sizeof(int) + 255) & ~(size_t)255;
    int* elist  = (int*)((char*)d_ws + off);

    const int* src = edge_index;
    const int* dst = edge_index + E;

    tga_tf_gemm<<<(N + 127) / 128, 256, 0, stream>>>(node_features, Wf, bf, tf, N);
    tga_scan   <<<B, 256, 0, stream>>>(src, dst, target_nodes, E, counts, elist);
    tga_attn   <<<B, 128, 0, stream>>>(tf, target_nodes, src, dst, edge_ts,
                                       Wt1, bt1, Wt2, bt2,
                                       W_in,                 b_in,
                                       W_in + N_HID * N_HID, b_in + N_HID,
                                       W_in + 2 * N_HID * N_HID, b_in + 2 * N_HID,
                                       W_out, b_out, Wo, bo,
                                       counts, elist, (float*)d_out);
}